// DistanceProbe_64080912056509
// MI455X (gfx1250) — compile-verified
//
#include <hip/hip_runtime.h>

typedef __attribute__((ext_vector_type(2))) float v2f;
typedef __attribute__((ext_vector_type(8))) float v8f;

#define MODEL_DIM 1024
#define RANK      128
#define SEQ       512
#define NBATCH    8
#define ROWS      (NBATCH * SEQ)   // 4096 total rows of T

// ---------------------------------------------------------------------------
// Kernel 1: T = batch @ proj   ([4096,1024] x [1024,128] -> [4096,128])
// One wave computes one 16x16 tile of T with V_WMMA_F32_16X16X4_F32,
// stepping K by 4 (256 WMMA ops per tile).
//
// f32 WMMA fragment layout (ISA 7.12.2):
//   A (16x4, MxK): lanes 0-15 -> M=lane, VGPR{0,1}=K{0,1}; lanes 16-31 -> K{2,3}
//   B (4x16, KxN): lanes 0-15 -> N=lane, VGPR{0,1}=K{0,1}; lanes 16-31 -> K{2,3}
//   C/D (16x16):   VGPR r: lanes 0-15 -> M=r, lanes 16-31 -> M=r+8; N=lane&15
// ---------------------------------------------------------------------------
__global__ __launch_bounds__(256) void proj_gemm_wmma(
    const float* __restrict__ batch, const float* __restrict__ proj,
    float* __restrict__ T) {
  const int tid  = threadIdx.x;
  const int lane = tid & 31;
  const int half = lane >> 4;     // 0: K pair {0,1}, 1: K pair {2,3}
  const int l16  = lane & 15;
  const int widx = blockIdx.x * 8 + (tid >> 5);   // 2048 waves
  const int m0   = (widx >> 3) * 16;              // 256 row tiles
  const int n0   = (widx & 7) * 16;               // 8 col tiles

  const float* arow = batch + (size_t)(m0 + l16) * MODEL_DIM + 2 * half;
  const float* bcol = proj + (size_t)(2 * half) * RANK + n0 + l16;

  v8f acc = {};
#pragma unroll 4
  for (int kk = 0; kk < MODEL_DIM; kk += 4) {
    v2f a = *(const v2f*)(arow + kk);             // K=2*half+{0,1}, 8B aligned
    v2f b;
    b.x = bcol[(size_t)kk * RANK];                // proj[kk+2*half+0][n0+l16]
    b.y = bcol[(size_t)kk * RANK + RANK];         // proj[kk+2*half+1][n0+l16]
    acc = __builtin_amdgcn_wmma_f32_16x16x4_f32(
        false, a, false, b, (short)0, acc, false, false);
  }

  float* drow = T + (size_t)(m0 + 8 * half) * RANK + n0 + l16;
#pragma unroll
  for (int r = 0; r < 8; ++r)
    drow[(size_t)r * RANK] = acc[r];
}

// ---------------------------------------------------------------------------
// Kernel 2: norms[i] = ||T[i]||^2 ; one wave per row, float4 loads + shfl.
// ---------------------------------------------------------------------------
__global__ __launch_bounds__(256) void row_norms(
    const float* __restrict__ T, float* __restrict__ norms) {
  const int tid  = threadIdx.x;
  const int lane = tid & 31;
  const int row  = blockIdx.x * 8 + (tid >> 5);   // 4096 waves
  const float4 v = *(const float4*)(T + (size_t)row * RANK + lane * 4);
  float s = v.x * v.x + v.y * v.y + v.z * v.z + v.w * v.w;
#pragma unroll
  for (int off = 16; off > 0; off >>= 1)
    s += __shfl_xor(s, off, 32);
  if (lane == 0) norms[row] = s;
}

// ---------------------------------------------------------------------------
// Kernel 3: per batch b, G = T_b @ T_b^T via WMMA; out = n_i + n_j - 2*G.
// Since B = A^T, the B-fragment of T^T loads as plain rows of T (same
// addressing pattern as A, different row base) -- no transpose needed.
// ---------------------------------------------------------------------------
__global__ __launch_bounds__(256) void gram_dist_wmma(
    const float* __restrict__ T, const float* __restrict__ norms,
    float* __restrict__ out) {
  const int tid  = threadIdx.x;
  const int lane = tid & 31;
  const int half = lane >> 4;
  const int l16  = lane & 15;
  const int widx = blockIdx.x * 8 + (tid >> 5);   // 8192 waves
  const int b    = widx >> 10;                    // 1024 tiles per batch
  const int t    = widx & 1023;
  const int i0   = (t >> 5) * 16;
  const int j0   = (t & 31) * 16;

  const float* Tb = T + (size_t)b * SEQ * RANK;
  const float* ai = Tb + (size_t)(i0 + l16) * RANK + 2 * half;
  const float* aj = Tb + (size_t)(j0 + l16) * RANK + 2 * half;

  v8f acc = {};
#pragma unroll
  for (int kk = 0; kk < RANK; kk += 4) {
    v2f a  = *(const v2f*)(ai + kk);
    v2f bb = *(const v2f*)(aj + kk);   // B[k][n] = T_b[j0+n][k] fragment
    acc = __builtin_amdgcn_wmma_f32_16x16x4_f32(
        false, a, false, bb, (short)0, acc, false, false);
  }

  const float* nb  = norms + (size_t)b * SEQ;
  const float  nj  = nb[j0 + l16];
  float* orow = out + ((size_t)b * SEQ + (i0 + 8 * half)) * SEQ + j0 + l16;
#pragma unroll
  for (int r = 0; r < 8; ++r) {
    const int i = i0 + 8 * half + r;
    float v = nb[i] + nj - 2.0f * acc[r];
    v = (i == (j0 + l16)) ? 0.0f : fmaxf(v, 0.0f);  // exact-zero diagonal
    orow[(size_t)r * SEQ] = v;
  }
}

// ---------------------------------------------------------------------------
extern "C" void kernel_launch(void* const* d_in, const int* in_sizes, int n_in,
                              void* d_out, int out_size, void* d_ws, size_t ws_size,
                              hipStream_t stream) {
  const float* batch = (const float*)d_in[0];   // [8,512,1024]
  const float* proj  = (const float*)d_in[1];   // [1024,128]
  float* out = (float*)d_out;                   // [8,512,512]

  float* T     = (float*)d_ws;                  // 4096*128 f32 = 2 MB
  float* norms = T + (size_t)ROWS * RANK;       // 4096 f32

  proj_gemm_wmma<<<dim3(256),  dim3(256), 0, stream>>>(batch, proj, T);
  row_norms     <<<dim3(512),  dim3(256), 0, stream>>>(T, norms);
  gram_dist_wmma<<<dim3(1024), dim3(256), 0, stream>>>(T, norms, out);
}